// GNN_19052474925086
// MI455X (gfx1250) — compile-verified
//
#include <hip/hip_runtime.h>
#include <hip/hip_bf16.h>

typedef __attribute__((ext_vector_type(2))) float v2f;
typedef __attribute__((ext_vector_type(8))) float v8f;

#define NNODES 100000
#define NEDGES 400000
#define DIM    300
#define NLAYERS 5
#define BN_EPS 1e-5f

// ---------------------------------------------------------------------------
// Initial node embedding: h[i,:] = x_emb1[x[i,0],:] + x_emb2[x[i,1],:]
// ---------------------------------------------------------------------------
__global__ void init_h_kernel(const int* __restrict__ x,
                              const float* __restrict__ e1,   // [120, D]
                              const float* __restrict__ e2,   // [3, D]
                              float* __restrict__ h)
{
    long long idx = (long long)blockIdx.x * blockDim.x + threadIdx.x;
    long long total = (long long)NNODES * DIM;
    if (idx >= total) return;
    int i = (int)(idx / DIM);
    int d = (int)(idx % DIM);
    int a = x[2 * i];
    int c = x[2 * i + 1];
    h[idx] = e1[(size_t)a * DIM + d] + e2[(size_t)c * DIM + d];
}

// ---------------------------------------------------------------------------
// Message gather + scatter-sum (incl. implicit self-loops with token 4).
// h fits in L2 (120MB < 192MB) so gathers/atomics are L2-resident.
// Each thread handles one (edge, float4-chunk) pair -> coalesced row reads.
// ---------------------------------------------------------------------------
__global__ void scatter_kernel(const float* __restrict__ h,
                               const int* __restrict__ edge_index, // [2, E]
                               const int* __restrict__ edge_attr,  // [E, 2]
                               const float* __restrict__ ee1,      // [6, D] (layer l)
                               const float* __restrict__ ee2,      // [3, D] (layer l)
                               float* __restrict__ agg)
{
    const int Q = DIM / 4; // 75 float4 chunks per row
    long long idx = (long long)blockIdx.x * blockDim.x + threadIdx.x;
    long long total = (long long)(NEDGES + NNODES) * Q;
    if (idx >= total) return;
    int e = (int)(idx / Q);
    int q = (int)(idx % Q);

    int src, dst, a0, a1;
    if (e < NEDGES) {
        src = edge_index[e];
        dst = edge_index[NEDGES + e];
        a0  = edge_attr[2 * e];
        a1  = edge_attr[2 * e + 1];
    } else {                 // self-loop
        src = dst = e - NEDGES;
        a0 = 4;              // SELF_LOOP_TOKEN
        a1 = 0;
    }

    const float4 hv = ((const float4*)(h   + (size_t)src * DIM))[q];
    const float4 v1 = ((const float4*)(ee1 + (size_t)a0  * DIM))[q];
    const float4 v2 = ((const float4*)(ee2 + (size_t)a1  * DIM))[q];

    float* out = agg + (size_t)dst * DIM + q * 4;
    atomicAdd(out + 0, hv.x + v1.x + v2.x);
    atomicAdd(out + 1, hv.y + v1.y + v2.y);
    atomicAdd(out + 2, hv.z + v1.z + v2.z);
    atomicAdd(out + 3, hv.w + v1.w + v2.w);
}

// ---------------------------------------------------------------------------
// fp32 WMMA GEMM: C[M,Ncol] = optReLU(A[M,K] @ B[K,Ncol] + bias[Ncol])
// One wave owns a 64x32 tile: 4 M-subtiles x 2 N-subtiles = 8 v8f
// accumulators. Each k-step (K+=4): 4 A-frag b64 loads + 4 B b32 loads feed
// 8 V_WMMA_F32_16X16X4_F32 — A fragments reused across both N-subtiles.
// Fragment layouts per ISA 7.12.2:
//   A 16x4 : lane l(0-15) row=l, holds K = {2*half, 2*half+1} in {v0,v1}
//   B 4x16 : lane col = l&15,   holds K = {2*half, 2*half+1} in {v0,v1}
//   C/D    : v8f, element g -> row g + 8*half, col = l&15
// K (300 or 600) is a multiple of 4; ragged M/N edges use clamped loads +
// masked stores so EXEC stays all-ones around every WMMA.
// ---------------------------------------------------------------------------
__global__ __launch_bounds__(128)
void gemm_wmma_f32(const float* __restrict__ A,
                   const float* __restrict__ B,
                   const float* __restrict__ bias,
                   float* __restrict__ C,
                   int M, int K, int Ncol, int relu)
{
    int w    = blockIdx.x * 4 + (threadIdx.x >> 5); // 4 waves / block
    int numN = (Ncol + 31) >> 5;                    // 32-wide column tiles
    int numM = (M + 63) >> 6;                       // 64-tall row tiles
    if (w >= numM * numN) return;                   // wave-uniform exit

    int mt   = w / numN;
    int nt   = w % numN;
    int lane = threadIdx.x & 31;
    int half = lane >> 4;                       // 0 or 1
    int l16  = lane & 15;
    int kof  = half * 2;

    int col0  = nt * 32 + l16;
    int col1  = col0 + 16;
    int colc0 = (col0 < Ncol) ? col0 : (Ncol - 1); // clamp (masked on store)
    int colc1 = (col1 < Ncol) ? col1 : (Ncol - 1);

    // Row pointers for the 4 M-subtiles (clamped rows are masked on store).
    const float* ap[4];
#pragma unroll
    for (int s = 0; s < 4; ++s) {
        int rr = mt * 64 + s * 16 + l16;
        int rc = (rr < M) ? rr : (M - 1);
        ap[s] = A + (size_t)rc * K + kof;
    }
    // B pointers: rows (k+kof) and (k+kof+1), advanced by 4*Ncol per k-step.
    const float* bp0a = B + (size_t)kof * Ncol + colc0;
    const float* bp0b = bp0a + Ncol;
    const float* bp1a = B + (size_t)kof * Ncol + colc1;
    const float* bp1b = bp1a + Ncol;

    v8f acc[8] = {};
    for (int k = 0; k < K; k += 4) {
        v2f b0, b1;
        b0.x = bp0a[0]; b0.y = bp0b[0];
        b1.x = bp1a[0]; b1.y = bp1b[0];
        bp0a += 4 * Ncol; bp0b += 4 * Ncol;
        bp1a += 4 * Ncol; bp1b += 4 * Ncol;
#pragma unroll
        for (int s = 0; s < 4; ++s) {
            v2f a = *(const v2f*)(ap[s] + k);
            acc[s]     = __builtin_amdgcn_wmma_f32_16x16x4_f32(
                false, a, false, b0, (short)0, acc[s],     false, false);
            acc[s + 4] = __builtin_amdgcn_wmma_f32_16x16x4_f32(
                false, a, false, b1, (short)0, acc[s + 4], false, false);
        }
    }

    float bb0 = bias[colc0];
    float bb1 = bias[colc1];
#pragma unroll
    for (int s = 0; s < 4; ++s) {
#pragma unroll
        for (int g = 0; g < 8; ++g) {
            int row = mt * 64 + s * 16 + g + half * 8;
            if (row < M) {
                if (col0 < Ncol) {
                    float v = acc[s][g] + bb0;
                    if (relu) v = v > 0.0f ? v : 0.0f;
                    C[(size_t)row * Ncol + col0] = v;
                }
                if (col1 < Ncol) {
                    float v = acc[s + 4][g] + bb1;
                    if (relu) v = v > 0.0f ? v : 0.0f;
                    C[(size_t)row * Ncol + col1] = v;
                }
            }
        }
    }
}

// ---------------------------------------------------------------------------
// BatchNorm batch statistics: per-column sum and sum-of-squares.
// Threads map to columns (coalesced row reads), grid.y splits rows.
// ---------------------------------------------------------------------------
__global__ void bn_stats_kernel(const float* __restrict__ H,
                                float* __restrict__ sums,
                                float* __restrict__ sumsq,
                                int M, int Dd)
{
    int d = blockIdx.x * blockDim.x + threadIdx.x;
    if (d >= Dd) return;
    int chunk = (M + gridDim.y - 1) / gridDim.y;
    int r0 = blockIdx.y * chunk;
    int r1 = r0 + chunk; if (r1 > M) r1 = M;
    float s = 0.0f, s2 = 0.0f;
    for (int rr = r0; rr < r1; ++rr) {
        float v = H[(size_t)rr * Dd + d];
        s  += v;
        s2 += v * v;
    }
    atomicAdd(sums  + d, s);
    atomicAdd(sumsq + d, s2);
}

// ---------------------------------------------------------------------------
// BatchNorm apply (+ optional ReLU): biased variance, training mode.
// ---------------------------------------------------------------------------
__global__ void bn_apply_kernel(const float* __restrict__ H,
                                const float* __restrict__ sums,
                                const float* __restrict__ sumsq,
                                const float* __restrict__ gamma,
                                const float* __restrict__ beta,
                                float* __restrict__ Out,
                                int M, int Dd, int relu)
{
    long long idx = (long long)blockIdx.x * blockDim.x + threadIdx.x;
    long long total = (long long)M * Dd;
    if (idx >= total) return;
    int d = (int)(idx % Dd);
    float invM = 1.0f / (float)M;
    float mean = sums[d] * invM;
    float var  = sumsq[d] * invM - mean * mean;
    float rs   = rsqrtf(var + BN_EPS);
    float v    = (H[idx] - mean) * rs * gamma[d] + beta[d];
    if (relu) v = fmaxf(v, 0.0f);
    Out[idx] = v;
}

// ---------------------------------------------------------------------------
// Host launcher
// ---------------------------------------------------------------------------
extern "C" void kernel_launch(void* const* d_in, const int* in_sizes, int n_in,
                              void* d_out, int out_size, void* d_ws, size_t ws_size,
                              hipStream_t stream) {
    (void)in_sizes; (void)n_in; (void)out_size; (void)ws_size;
    const int N = NNODES, E = NEDGES, D = DIM, L = NLAYERS;

    const int*   x          = (const int*)  d_in[0];
    const int*   edge_index = (const int*)  d_in[1];
    const int*   edge_attr  = (const int*)  d_in[2];
    const float* x_emb1     = (const float*)d_in[3];
    const float* x_emb2     = (const float*)d_in[4];
    const float* edge_emb1  = (const float*)d_in[5];  // [L,6,D]
    const float* edge_emb2  = (const float*)d_in[6];  // [L,3,D]
    const float* W1         = (const float*)d_in[7];  // [L,D,2D]
    const float* b1         = (const float*)d_in[8];  // [L,2D]
    const float* W2         = (const float*)d_in[9];  // [L,2D,D]
    const float* b2         = (const float*)d_in[10]; // [L,D]
    const float* gamma      = (const float*)d_in[11]; // [L,D]
    const float* beta       = (const float*)d_in[12]; // [L,D]

    float* h     = (float*)d_out;                 // [N,D] (final output lives here)
    float* agg   = (float*)d_ws;                  // [N,D]   (also reused as h2)
    float* mid   = agg + (size_t)N * D;           // [N,2D]
    float* sums  = mid + (size_t)N * 2 * D;       // [D]
    float* sumsq = sums + D;                      // [D]

    // h0 = x_emb1[x[:,0]] + x_emb2[x[:,1]]
    {
        long long tot = (long long)N * D;
        init_h_kernel<<<(int)((tot + 255) / 256), 256, 0, stream>>>(x, x_emb1, x_emb2, h);
    }

    for (int l = 0; l < L; ++l) {
        // agg = 0
        hipMemsetAsync(agg, 0, (size_t)N * D * sizeof(float), stream);

        // agg[dst] += h[src] + edge_emb (incl. self-loops)
        {
            long long tot = (long long)(E + N) * (D / 4);
            scatter_kernel<<<(int)((tot + 255) / 256), 256, 0, stream>>>(
                h, edge_index, edge_attr,
                edge_emb1 + (size_t)l * 6 * D,
                edge_emb2 + (size_t)l * 3 * D,
                agg);
        }

        // mid = relu(agg @ W1 + b1)      [N,300]x[300,600]
        {
            int Ncol = 2 * D;
            int waves = ((Ncol + 31) / 32) * ((N + 63) / 64);
            gemm_wmma_f32<<<(waves + 3) / 4, 128, 0, stream>>>(
                agg, W1 + (size_t)l * D * 2 * D, b1 + (size_t)l * 2 * D,
                mid, N, D, Ncol, 1);
        }

        // h2(agg) = mid @ W2 + b2        [N,600]x[600,300]
        {
            int Ncol = D;
            int waves = ((Ncol + 31) / 32) * ((N + 63) / 64);
            gemm_wmma_f32<<<(waves + 3) / 4, 128, 0, stream>>>(
                mid, W2 + (size_t)l * 2 * D * D, b2 + (size_t)l * D,
                agg, N, 2 * D, Ncol, 0);
        }

        // BatchNorm (training mode, biased var) + ReLU except last layer
        hipMemsetAsync(sums, 0, 2 * (size_t)D * sizeof(float), stream);
        {
            dim3 grid((D + 127) / 128, 256);
            bn_stats_kernel<<<grid, 128, 0, stream>>>(agg, sums, sumsq, N, D);
        }
        {
            long long tot = (long long)N * D;
            bn_apply_kernel<<<(int)((tot + 255) / 256), 256, 0, stream>>>(
                agg, sums, sumsq,
                gamma + (size_t)l * D, beta + (size_t)l * D,
                h, N, D, (l != L - 1) ? 1 : 0);
        }
    }
}